// spiking_coESN_66915590472001
// MI455X (gfx1250) — compile-verified
//
#include <hip/hip_runtime.h>

// ---------------------------------------------------------------------------
// Spiking coESN on MI455X (gfx1250, wave32).
//
// Per step:  cur = x_t (x) x2h  +  s @ h2h  +  bias   (S(32x1024) W(1024x1024))
// then elementwise LIF + HRF updates; spikes feed the next step (serial scan).
//
//  * f32 WMMA (v_wmma_f32_16x16x4_f32): exact math (spikes in {0,1}, weights
//    need f32 precision for the thresholded recurrence).
//  * Each wave owns ONE 16-column N-tile and computes BOTH 16-row M-tiles:
//    one b64 B-fragment load feeds two WMMAs (2 independent accum chains,
//    4 MB/step total L2 B traffic, weights fully L2-resident).
//  * h2h pre-swizzled once into per-lane B-fragment order (coalesced b64).
//  * A fragments interleaved [kb][mt][lane] so both M-tile loads of a k-chunk
//    fold into ds_load_2addr_b64 immediate offsets (no VALU address math).
//  * Spikes exchanged via a packed A-fragment global double buffer; consumers
//    pull 128 KB/step into LDS with GLOBAL_LOAD_ASYNC_TO_LDS_B128 (ASYNCcnt,
//    no VGPR round-trip), producers scatter b32 stores.
//  * Persistent 8-WG grid; per-step arrive-counter grid barrier
//    (agent-scope __hip_atomic + s_sleep spin). All neuron state in VGPRs.
// ---------------------------------------------------------------------------

typedef __attribute__((ext_vector_type(2))) float v2f;
typedef __attribute__((ext_vector_type(8))) float v8f;

#define NHID   1024
#define SEQL   512
#define NBLK   8           // workgroups in persistent grid
#define KBLK   256         // 1024 / 4
#define DTC    0.042f
#define REF_DECAY 0.8453544f   // exp(-0.042/0.25)

// workspace layout (bytes):
//   [0,      4096)   per-step barrier counters (512 ints)
//   [4096, 266240)   packed spike double buffer: 2 x 32768 floats
//   [266240, +4MB)   packed weights (B fragments)
#define WS_SPIKE_OFF 4096
#define WS_WPACK_OFF 266240
#define SPIKE_FLOATS 32768   // 256 kb * 2 Mtiles * 32 lanes * 2

// Pack h2h[k][c] (row-major 1024x1024) into per-lane WMMA B fragments:
// wpack[((nt*256 + kb)*32 + lane)] = float2{ W[k0, c], W[k0+1, c] },
//   k0 = kb*4 + (lane>=16 ? 2 : 0),  c = nt*16 + (lane & 15)
__global__ __launch_bounds__(256) void pack_weights(
    const float* __restrict__ h2h, float* __restrict__ wpack) {
  int idx  = blockIdx.x * 256 + threadIdx.x;   // 0 .. 524287
  int lane = idx & 31;
  int kb   = (idx >> 5) & 255;
  int nt   = idx >> 13;
  int k0   = kb * 4 + ((lane >> 4) << 1);
  int c    = nt * 16 + (lane & 15);
  v2f val;
  val.x = h2h[(size_t)k0 * NHID + c];
  val.y = h2h[(size_t)(k0 + 1) * NHID + c];
  ((v2f*)wpack)[idx] = val;
}

__global__ __launch_bounds__(256) void esn_persistent(
    const float* __restrict__ x,      // (32, 512, 1)
    const float* __restrict__ x2h,    // (1, 1024)
    const float* __restrict__ bias,   // (1024)
    const float* __restrict__ gam,    // (1024)
    const float* __restrict__ eps,    // (1024)
    const float* __restrict__ wpack,  // packed B fragments
    float* __restrict__ spikebuf,     // 2 x 32768 floats, zero-initialized
    int*   __restrict__ bar,          // 512 ints, zeroed
    float* __restrict__ out)          // (32, 1024)
{
  extern __shared__ __align__(16) float smem[];
  // A fragments interleaved [kb][mt][lane] (float2 elems): both Mt tiles of a
  // k-block are 256B apart -> ds_load_2addr_b64 immediate offsets.
  float* sA = smem;                  // 32768 floats
  float* sX = smem + SPIKE_FLOATS;   // 32 floats: x[:, t]

  const int tid  = threadIdx.x;
  const int lane = tid & 31;
  const int wv   = tid >> 5;                 // 0..7
  const int nt   = blockIdx.x * 8 + wv;      // column tile 0..63 (one per wave)
  const int c    = nt * 16 + (lane & 15);    // owned column
  const int hi   = (lane >> 4) & 1;

  const float bias_c = bias[c];
  const float g_c    = gam[c];
  const float e_c    = eps[c];
  const float xw_c   = x2h[c];

  // fragment base pointers
  const v2f* Bf  = (const v2f*)wpack + (size_t)nt * (KBLK * 32) + lane;
  const v2f* Af0 = (const v2f*)sA + lane;        // [kb][0][lane], stride 64
  const v2f* Af1 = Af0 + 32;                     // [kb][1][lane]

  // spike scatter base for column c (element c&1 of the packed float2):
  // offset(mt,v) = ((kb2*2 + mt)*32 + lane2 + v)*2 + (c&1)
  const int kb2   = c >> 2;
  const int lane2 = ((c >> 1) & 1) * 16 + hi * 8;
  const int sbase = (kb2 * 64 + lane2) * 2 + (c & 1);

  v8f lifv[2] = {{}, {}}, hy[2] = {{}, {}}, hz[2] = {{}, {}};
  v8f ref[2]  = {{}, {}}, rate[2] = {{}, {}};

  // generic shared pointers carry the LDS byte address in their low 32 bits
  const unsigned ldsA = (unsigned)(size_t)sA;

  for (int t = 0; t < SEQL; ++t) {
    const int par = t & 1;

    // --- async-copy previous spikes (packed A fragments) into LDS ---------
    const float4* src = (const float4*)(spikebuf + par * SPIKE_FLOATS);
    #pragma unroll 4
    for (int i = tid; i < SPIKE_FLOATS / 4; i += 256) {
      asm volatile("global_load_async_to_lds_b128 %0, %1, off"
                   :: "v"(ldsA + (unsigned)i * 16u), "v"(src + i)
                   : "memory");
    }
    if (tid < 32) sX[tid] = x[tid * SEQL + t];
    asm volatile("s_wait_asynccnt 0x0" ::: "memory");
    __syncthreads();

    // --- init accumulators: bias + rank-1 input drive ---------------------
    v8f acc0, acc1;
    #pragma unroll
    for (int v = 0; v < 8; ++v) {
      acc0[v] = fmaf(sX[hi * 8 + v],      xw_c, bias_c);
      acc1[v] = fmaf(sX[16 + hi * 8 + v], xw_c, bias_c);
    }

    // --- GEMM: 512 x v_wmma_f32_16x16x4_f32 (B reused by both M-tiles) ----
    #pragma unroll 1
    for (int kc = 0; kc < 4; ++kc) {
      __builtin_prefetch((const void*)(Bf + (size_t)(kc + 1) * 64 * 32), 0, 1);
      const int k0 = kc * 64;
      #pragma unroll 4
      for (int kb = k0; kb < k0 + 64; ++kb) {
        v2f b  = Bf[(size_t)kb * 32];
        v2f a0 = Af0[(size_t)kb * 64];
        v2f a1 = Af1[(size_t)kb * 64];
        acc0 = __builtin_amdgcn_wmma_f32_16x16x4_f32(
            false, a0, false, b, (short)0, acc0, false, false);
        acc1 = __builtin_amdgcn_wmma_f32_16x16x4_f32(
            false, a1, false, b, (short)0, acc1, false, false);
      }
    }

    // --- elementwise neuron update (16 neurons per lane) ------------------
    float* dst = spikebuf + (par ^ 1) * SPIKE_FLOATS + sbase;
    #pragma unroll
    for (int mt = 0; mt < 2; ++mt) {
      v8f acc = mt ? acc1 : acc0;
      #pragma unroll
      for (int v = 0; v < 8; ++v) {
        float cur = acc[v];
        float lv  = lifv[mt][v];
        lv = lv + DTC * (cur - lv * 0.05f);          // -lv/tau_m + cur
        float ls = (lv > 0.05f) ? 1.0f : 0.0f;
        lv -= ls * 0.05f;                            // subtractive reset
        lifv[mt][v] = lv;
        float z = hz[mt][v] + DTC * (35.0f * ls - g_c * hy[mt][v] - e_c * hz[mt][v]);
        float y = hy[mt][v] + DTC * z;
        hz[mt][v] = z; hy[mt][v] = y;
        float sn = ((y - 0.05f - ref[mt][v]) > 0.0f) ? 1.0f : 0.0f;
        ref[mt][v] = ref[mt][v] * REF_DECAY + sn;
        rate[mt][v] += sn;
        // scatter into packed interleaved A-fragment layout (other parity)
        dst[mt * 64 + 2 * v] = sn;
      }
    }

    // --- grid-wide step barrier (release, arrive, spin, acquire) ----------
    __threadfence();
    __syncthreads();
    if (tid == 0) {
      __hip_atomic_fetch_add(bar + t, 1, __ATOMIC_ACQ_REL,
                             __HIP_MEMORY_SCOPE_AGENT);
      while (__hip_atomic_load(bar + t, __ATOMIC_ACQUIRE,
                               __HIP_MEMORY_SCOPE_AGENT) < NBLK)
        __builtin_amdgcn_s_sleep(1);
    }
    __syncthreads();
    __threadfence();
  }

  // --- mean firing rate ---------------------------------------------------
  const float inv = 1.0f / (float)SEQL;
  #pragma unroll
  for (int mt = 0; mt < 2; ++mt)
    #pragma unroll
    for (int v = 0; v < 8; ++v)
      out[(size_t)(mt * 16 + hi * 8 + v) * NHID + c] = rate[mt][v] * inv;
}

extern "C" void kernel_launch(void* const* d_in, const int* in_sizes, int n_in,
                              void* d_out, int out_size, void* d_ws, size_t ws_size,
                              hipStream_t stream) {
  const float* x    = (const float*)d_in[0];   // (32,512,1)
  const float* x2h  = (const float*)d_in[1];   // (1,1024)
  const float* h2h  = (const float*)d_in[2];   // (1024,1024)
  const float* bias = (const float*)d_in[3];   // (1024)
  const float* gam  = (const float*)d_in[4];   // (1024)
  const float* eps  = (const float*)d_in[5];   // (1024)

  char* ws = (char*)d_ws;
  int*   bar      = (int*)ws;
  float* spikebuf = (float*)(ws + WS_SPIKE_OFF);
  float* wpack    = (float*)(ws + WS_WPACK_OFF);

  // zero barriers + both spike parity buffers (s_0 = 0)
  hipMemsetAsync(ws, 0, WS_WPACK_OFF, stream);

  // one-time weight swizzle into WMMA B-fragment order (4 MB)
  pack_weights<<<2048, 256, 0, stream>>>(h2h, wpack);

  // persistent scan: 8 WGs x 256 threads, 128KB+ dynamic LDS
  esn_persistent<<<NBLK, 256, (SPIKE_FLOATS + 32) * sizeof(float), stream>>>(
      x, x2h, bias, gam, eps, wpack, spikebuf, bar, (float*)d_out);
}